// Set2GraphModel_2834678415691
// MI455X (gfx1250) — compile-verified
//
#include <hip/hip_runtime.h>
#include <math.h>

typedef __attribute__((ext_vector_type(2))) float v2f;
typedef __attribute__((ext_vector_type(8))) float v8f;

#define HID   256
#define NPTS  128
#define BATCH 16
#define NPAIR (NPTS * (NPTS - 1))   // 16256

// LDS row stride padding for the node kernel tiles
#define LPAD 8
#define LSTR (HID + LPAD)           // 264 floats per row

// ---------------------------------------------------------------------------
// Node kernel: one block per 16-row tile of the flattened [B*N, ...] rows.
// Fused:  H1 = relu(x@W1+b1)  ->  emb = H1@W2+b2  ->  U = emb@W3_top+b3,
//                                                     V = emb@W3_bot
// H1 and emb live only in LDS; U/V go to workspace.
// All matmuls use V_WMMA_F32_16X16X4_F32 (fp32 in, fp32 acc).
// ---------------------------------------------------------------------------
__global__ __launch_bounds__(256) void s2g_node_kernel(
    const float* __restrict__ x,    // [2048, 16]
    const float* __restrict__ W1,   // [16, 256]
    const float* __restrict__ b1,   // [256]
    const float* __restrict__ W2,   // [256, 256]
    const float* __restrict__ b2,   // [256]
    const float* __restrict__ W3,   // [512, 256]  rows 0..255 = top, 256..511 = bottom
    const float* __restrict__ b3,   // [256]
    float* __restrict__ U,          // [2048, 256]
    float* __restrict__ V)          // [2048, 256]
{
    __shared__ float H1[16 * LSTR];
    __shared__ float EM[16 * LSTR];

    const int tid  = threadIdx.x;
    const int lane = tid & 31;
    const int wave = tid >> 5;      // 0..7
    const int m    = lane & 15;     // row (A) / col (B,C,D) within tile
    const int kh   = lane >> 4;     // K-half selector for A/B fragments
    const int row0 = blockIdx.x * 16;

    // ---- Stage A: H1 = relu(x @ W1 + b1); 16 column tiles, 2 per wave ----
    #pragma unroll
    for (int q = 0; q < 2; ++q) {
        const int n0 = (wave * 2 + q) * 16;
        const float bv = b1[n0 + m];
        v8f acc;
        #pragma unroll
        for (int r = 0; r < 8; ++r) acc[r] = bv;
        #pragma unroll
        for (int k = 0; k < 16; k += 4) {
            const int ka = k + 2 * kh;
            v2f a, b;
            a.x = x[(row0 + m) * 16 + ka];
            a.y = x[(row0 + m) * 16 + ka + 1];
            b.x = W1[(ka    ) * HID + n0 + m];
            b.y = W1[(ka + 1) * HID + n0 + m];
            acc = __builtin_amdgcn_wmma_f32_16x16x4_f32(
                false, a, false, b, (short)0, acc, false, false);
        }
        #pragma unroll
        for (int r = 0; r < 8; ++r) {
            float v = acc[r];
            H1[(r + 8 * kh) * LSTR + n0 + m] = v > 0.f ? v : 0.f;
        }
    }
    __syncthreads();

    // ---- Stage B: emb = H1 @ W2 + b2; K = 256 from LDS ----
    #pragma unroll
    for (int q = 0; q < 2; ++q) {
        const int n0 = (wave * 2 + q) * 16;
        const float bv = b2[n0 + m];
        v8f acc;
        #pragma unroll
        for (int r = 0; r < 8; ++r) acc[r] = bv;
        #pragma unroll 4
        for (int k = 0; k < HID; k += 4) {
            const int ka = k + 2 * kh;
            v2f a, b;
            a.x = H1[m * LSTR + ka];
            a.y = H1[m * LSTR + ka + 1];
            b.x = W2[(ka    ) * HID + n0 + m];
            b.y = W2[(ka + 1) * HID + n0 + m];
            acc = __builtin_amdgcn_wmma_f32_16x16x4_f32(
                false, a, false, b, (short)0, acc, false, false);
        }
        #pragma unroll
        for (int r = 0; r < 8; ++r)
            EM[(r + 8 * kh) * LSTR + n0 + m] = acc[r];
    }
    __syncthreads();

    // ---- Stage C: U = emb@W3_top + b3 ; V = emb@W3_bot ; 32 tiles, 4/wave ----
    #pragma unroll
    for (int q = 0; q < 4; ++q) {
        const int tt  = wave * 4 + q;        // uniform per wave
        const bool isU = (tt < 16);
        const int n0  = (isU ? tt : tt - 16) * 16;
        const float bv = isU ? b3[n0 + m] : 0.f;
        const float* Bp = isU ? W3 : (W3 + 256 * HID);
        float* dst = isU ? U : V;
        v8f acc;
        #pragma unroll
        for (int r = 0; r < 8; ++r) acc[r] = bv;
        #pragma unroll 4
        for (int k = 0; k < HID; k += 4) {
            const int ka = k + 2 * kh;
            v2f a, b;
            a.x = EM[m * LSTR + ka];
            a.y = EM[m * LSTR + ka + 1];
            b.x = Bp[(ka    ) * HID + n0 + m];
            b.y = Bp[(ka + 1) * HID + n0 + m];
            acc = __builtin_amdgcn_wmma_f32_16x16x4_f32(
                false, a, false, b, (short)0, acc, false, false);
        }
        #pragma unroll
        for (int r = 0; r < 8; ++r)
            dst[(size_t)(row0 + r + 8 * kh) * HID + n0 + m] = acc[r];
    }
}

// ---------------------------------------------------------------------------
// Pair kernel: one block per (batch, 16x16 tile of (i,j) pairs).
// U/V tiles staged in LDS (padded stride 257 -> conflict-free), then each
// thread computes one pair: sigmoid(W4 . relu(U_i + V_j) + b4).
// ---------------------------------------------------------------------------
__global__ __launch_bounds__(256) void s2g_pair_kernel(
    const float* __restrict__ U,
    const float* __restrict__ V,
    const float* __restrict__ W4,   // [256]
    const float* __restrict__ b4,   // [1]
    float* __restrict__ out)        // [B, 16256]
{
    __shared__ float Us[16][HID + 1];
    __shared__ float Vs[16][HID + 1];
    __shared__ float W4s[HID];

    const int b   = blockIdx.y;
    const int i0  = (blockIdx.x >> 3) * 16;
    const int j0  = (blockIdx.x & 7) * 16;
    const int tid = threadIdx.x;

    for (int idx = tid; idx < 16 * HID; idx += 256) {
        const int r = idx >> 8;
        const int c = idx & (HID - 1);
        Us[r][c] = U[(size_t)(b * NPTS + i0 + r) * HID + c];
        Vs[r][c] = V[(size_t)(b * NPTS + j0 + r) * HID + c];
    }
    if (tid < HID) W4s[tid] = W4[tid];
    __syncthreads();

    const int li = tid >> 4;
    const int lj = tid & 15;
    const int i  = i0 + li;
    const int j  = j0 + lj;
    if (i == j) return;                      // diagonal pairs don't exist

    float acc = b4[0];
    #pragma unroll 8
    for (int k = 0; k < HID; ++k) {
        float s = Us[li][k] + Vs[lj][k];
        s = s > 0.f ? s : 0.f;
        acc = fmaf(s, W4s[k], acc);
    }
    const float sig = 1.f / (1.f + __expf(-acc));
    const int p = i * (NPTS - 1) + j - (j > i ? 1 : 0);   // row-major, skip diag
    out[(size_t)b * NPAIR + p] = sig;
}

// ---------------------------------------------------------------------------
extern "C" void kernel_launch(void* const* d_in, const int* in_sizes, int n_in,
                              void* d_out, int out_size, void* d_ws, size_t ws_size,
                              hipStream_t stream) {
    (void)in_sizes; (void)n_in; (void)out_size; (void)ws_size;
    const float* x  = (const float*)d_in[0];
    const float* W1 = (const float*)d_in[1];
    const float* b1 = (const float*)d_in[2];
    const float* W2 = (const float*)d_in[3];
    const float* b2 = (const float*)d_in[4];
    const float* W3 = (const float*)d_in[5];
    const float* b3 = (const float*)d_in[6];
    const float* W4 = (const float*)d_in[7];
    const float* b4 = (const float*)d_in[8];
    float* out = (float*)d_out;

    float* U = (float*)d_ws;                               // [2048,256] = 2 MB
    float* V = U + (size_t)BATCH * NPTS * HID;             // [2048,256] = 2 MB

    s2g_node_kernel<<<(BATCH * NPTS) / 16, 256, 0, stream>>>(
        x, W1, b1, W2, b2, W3, b3, U, V);

    dim3 grid(64, BATCH);                                  // 8x8 pair tiles x batch
    s2g_pair_kernel<<<grid, 256, 0, stream>>>(U, V, W4, b4, out);
}